// Wasserstein_reg_84507776516144
// MI455X (gfx1250) — compile-verified
//
#include <hip/hip_runtime.h>

// ---------------------------------------------------------------------------
// Sinkhorn-regularized Wasserstein (reference: N=40000, TOP=0.3, LAM=10,
// ITS=10, P=0.5).  m0 = m1 = 6000 after top-30% selection of each half.
// K is never materialized: every pass recomputes exp(-lam*|y1_i - y0_j|)
// on the fly (v_exp_f32 bound; working set ~50KB, L2/LDS resident).
// CDNA5 paths used: global_load_async_to_lds_b32 staging + s_wait_asynccnt,
// LDS broadcast reads, 320KB-LDS bitonic sort.  WMMA is not applicable
// (sequentially dependent matvecs with per-element transcendental).
// ---------------------------------------------------------------------------

#define M0 6000
#define M0P 6016              // padded stride
#define NSEG 16
#define ISEG 375              // 6000 / 16
#define JBLK 24               // 24 * 256 = 6144 >= 6000
#define A0 2.5e-5f            // P/nt = (1-P)/nc = 0.5/20000
#define SL 0.5f               // slack mass
#define LOG2E 1.4426950408889634f
#define NEG_FLT_MAX (-3.402823466e38f)

__device__ __forceinline__ float block_reduce(float v, float* sbuf, int tid, int n) {
  sbuf[tid] = v;
  __syncthreads();
  for (int s = n >> 1; s > 0; s >>= 1) {
    if (tid < s) sbuf[tid] += sbuf[tid + s];
    __syncthreads();
  }
  return sbuf[0];
}

// Stage (xi[i0+t], wi[i0+t]) pairs into LDS float2 via CDNA5 async-to-LDS.
__device__ __forceinline__ void stage_pairs_async(float2* sh, const float* xi,
                                                  const float* wi, int i0, int tid) {
  if (tid < ISEG) {
    unsigned lo = (unsigned)(size_t)(&sh[tid]);           // LDS byte offset
    unsigned long long ga = (unsigned long long)(size_t)(xi + i0 + tid);
    unsigned long long gb = (unsigned long long)(size_t)(wi + i0 + tid);
    asm volatile("global_load_async_to_lds_b32 %0, %1, off"
                 :: "v"(lo), "v"(ga) : "memory");
    asm volatile("global_load_async_to_lds_b32 %0, %1, off"
                 :: "v"(lo + 4u), "v"(gb) : "memory");
  }
  asm volatile("s_wait_asynccnt 0x0" ::: "memory");
  __syncthreads();
}

// -------------------- 1) bitonic sort of each half, keep top 6000 ----------
__global__ void sort_topk_kernel(const float* __restrict__ y_pred,
                                 float* __restrict__ ws) {
  extern __shared__ float smem[];                   // 32768 floats = 128KB
  const int NP = 32768, NH = 20000;
  const int tid = threadIdx.x;
  const float* src = y_pred + blockIdx.x * NH;
  for (int i = tid; i < NP; i += 1024)
    smem[i] = (i < NH) ? src[i] : NEG_FLT_MAX;
  __syncthreads();
  for (int k = 2; k <= NP; k <<= 1) {
    for (int jj = k >> 1; jj > 0; jj >>= 1) {
      for (int i = tid; i < NP; i += 1024) {
        int l = i ^ jj;
        if (l > i) {
          float a = smem[i], b = smem[l];
          bool up = ((i & k) == 0);
          if (up ? (a > b) : (a < b)) { smem[i] = b; smem[l] = a; }
        }
      }
      __syncthreads();
    }
  }
  // ascending sort, -inf padding at front -> top 6000 at the tail
  float* dst = ws + blockIdx.x * M0P;               // block0 -> y0s, block1 -> y1s
  for (int i = tid; i < M0; i += 1024) dst[i] = smem[NP - M0 + i];
}

// -------------------- 2) mean(M) partial sums ------------------------------
__global__ void meanm_kernel(const float* __restrict__ y0s,
                             const float* __restrict__ y1s,
                             float* __restrict__ mpart) {
  __shared__ float s1[512];
  __shared__ float sred[256];
  const int tid = threadIdx.x;
  const int j = blockIdx.x * 256 + tid;
  const int i0 = blockIdx.y * 500;
  for (int i = tid; i < 500; i += 256) s1[i] = y1s[i0 + i];
  __syncthreads();
  const float xj = (j < M0) ? y0s[j] : 0.0f;
  float acc = 0.0f;
  for (int ii = 0; ii < 500; ++ii) acc += fabsf(s1[ii] - xj);
  if (j >= M0) acc = 0.0f;
  float tot = block_reduce(acc, sred, tid, 256);
  if (tid == 0) mpart[blockIdx.y * JBLK + blockIdx.x] = tot;
}

// -------------------- 3) scalar init: lam, delta, kd; u = a ----------------
__global__ void init_kernel(const float* __restrict__ mpart,
                            const float* __restrict__ y0s,
                            const float* __restrict__ y1s,
                            float* __restrict__ u, float* __restrict__ scal) {
  __shared__ float sred[512];
  const int tid = threadIdx.x;
  float local = (tid < 288) ? mpart[tid] : 0.0f;
  float tot = block_reduce(local, sred, tid, 512);
  for (int i = tid; i < M0; i += 512) u[i] = A0;
  if (tid == 0) {
    u[M0] = SL;
    float meanM = tot / 36000000.0f;
    float lam = 10.0f / meanM;
    // arrays ascending: max at [5999], min at [0]
    float delta = fmaxf(y1s[M0 - 1] - y0s[0], y0s[M0 - 1] - y1s[0]);
    float kd = __builtin_amdgcn_exp2f(-lam * LOG2E * delta) + 1e-6f;
    scal[0] = -lam * LOG2E;   // c2: exp(-lam*d) = exp2(c2*d)
    scal[1] = delta;
    scal[2] = kd;
    scal[3] = 6000.0f * A0;   // S_u initial
    scal[5] = lam;
  }
}

// -------------------- 4) exp-matvec partial: out_j = sum_i exp()*w_i -------
__global__ void colpass_kernel(const float* __restrict__ xo,
                               const float* __restrict__ xi,
                               const float* __restrict__ wi,
                               float* __restrict__ part,
                               const float* __restrict__ scal) {
  __shared__ float2 sh[384];
  const int tid = threadIdx.x;
  const int j = blockIdx.x * 256 + tid;
  const int i0 = blockIdx.y * ISEG;
  const float c2 = scal[0];
  stage_pairs_async(sh, xi, wi, i0, tid);
  const float xj = (j < M0) ? xo[j] : 0.0f;
  float acc = 0.0f;
#pragma unroll 5
  for (int ii = 0; ii < ISEG; ++ii) {
    float2 p = sh[ii];
    acc += __builtin_amdgcn_exp2f(c2 * fabsf(p.x - xj)) * p.y;
  }
  if (j < M0) part[blockIdx.y * M0P + j] = acc;
}

// -------------------- 5) reduce partials, slack, vout = A0/t, sum ----------
__global__ void reduce_vec_kernel(const float* __restrict__ part,
                                  const float* __restrict__ vin,
                                  float* __restrict__ vout,
                                  float* __restrict__ scal,
                                  int in_idx, int out_idx) {
  __shared__ float sred[1024];
  const int tid = threadIdx.x;
  const float kd = scal[2];
  const float Sin = scal[in_idx];
  const float vsl = vin[M0];
  const float addc = 1e-6f * Sin + kd * vsl;   // +1e-6 term + slack column
  float local = 0.0f;
  for (int j = tid; j < M0; j += 1024) {
    float t = 0.0f;
#pragma unroll
    for (int s = 0; s < NSEG; ++s) t += part[s * M0P + j];
    t += addc;
    float o = A0 / t;
    vout[j] = o;
    local += o;
  }
  float tot = block_reduce(local, sred, tid, 1024);
  if (tid == 0) {
    float tsl = kd * Sin + (1.0f + 1e-6f) * vsl;  // slack-vs-slack corner
    vout[M0] = SL / tsl;
    scal[out_idx] = tot;
  }
}

// -------------------- 6) D main block: sum u_i v_j (exp+1e-6) d_ij ---------
__global__ void dpass_kernel(const float* __restrict__ y0s,
                             const float* __restrict__ y1s,
                             const float* __restrict__ u,
                             const float* __restrict__ v,
                             float* __restrict__ dpart,
                             const float* __restrict__ scal) {
  __shared__ float2 sh[384];
  __shared__ float sred[256];
  const int tid = threadIdx.x;
  const int j = blockIdx.x * 256 + tid;
  const int i0 = blockIdx.y * ISEG;
  const float c2 = scal[0];
  stage_pairs_async(sh, y1s, u, i0, tid);
  const float xj = (j < M0) ? y0s[j] : 0.0f;
  const float vj = (j < M0) ? v[j] : 0.0f;
  float acc = 0.0f;
#pragma unroll 5
  for (int ii = 0; ii < ISEG; ++ii) {
    float2 p = sh[ii];
    float d = fabsf(p.x - xj);
    acc += p.y * (__builtin_amdgcn_exp2f(c2 * d) + 1e-6f) * d;
  }
  acc *= vj;
  float tot = block_reduce(acc, sred, tid, 256);
  if (tid == 0) dpart[blockIdx.y * JBLK + blockIdx.x] = tot;
}

// -------------------- 7) finalize D ----------------------------------------
__global__ void dfinal_kernel(const float* __restrict__ dpart,
                              const float* __restrict__ scal,
                              const float* __restrict__ u,
                              const float* __restrict__ v,
                              float* __restrict__ out) {
  __shared__ float sred[512];
  const int tid = threadIdx.x;
  float local = (tid < JBLK * NSEG) ? dpart[tid] : 0.0f;
  float tot = block_reduce(local, sred, tid, 512);
  if (tid == 0) {
    float kd = scal[2], delta = scal[1], Su = scal[3], Sv = scal[4];
    // slack column: v_sl*kd*delta*Su ; slack row: u_sl*kd*delta*Sv ; corner Mt=0
    float D = 2.0f * (tot + kd * delta * (v[M0] * Su + u[M0] * Sv));
    out[0] = D; out[1] = 0.0f; out[2] = 0.0f; out[3] = 0.0f;
  }
}

extern "C" void kernel_launch(void* const* d_in, const int* in_sizes, int n_in,
                              void* d_out, int out_size, void* d_ws, size_t ws_size,
                              hipStream_t stream) {
  const float* y_pred = (const float*)d_in[0];
  float* ws = (float*)d_ws;

  // workspace layout (floats); total ~473 KB
  float* Y0    = ws;                       // 6016
  float* Y1    = ws + M0P;                 // 6016
  float* U     = ws + 2 * M0P;             // 6016 (6001 used)
  float* W     = ws + 3 * M0P;             // 6016 (6001 used)
  float* PART  = ws + 4 * M0P;             // 16 * 6016
  float* MPART = PART + NSEG * M0P;        // 288
  float* DPART = MPART + 320;              // 384
  float* SCAL  = DPART + 384;              // 16
  (void)in_sizes; (void)n_in; (void)out_size; (void)ws_size;

  // 1) top-6000 selection of each half (exact, in-LDS bitonic, 128KB dyn LDS)
  sort_topk_kernel<<<2, 1024, 32768 * sizeof(float), stream>>>(y_pred, ws);

  // 2-3) mean(M), delta, lam, kd; u = a
  meanm_kernel<<<dim3(JBLK, 12), 256, 0, stream>>>(Y0, Y1, MPART);
  init_kernel<<<1, 512, 0, stream>>>(MPART, Y0, Y1, U, SCAL);

  // 4) 10 Sinkhorn iterations (t = K^T u -> w = b/t ; r = K w -> u = a/r)
  const dim3 cgrid(JBLK, NSEG);
  for (int it = 0; it < 10; ++it) {
    colpass_kernel<<<cgrid, 256, 0, stream>>>(Y0, Y1, U, PART, SCAL);
    reduce_vec_kernel<<<1, 1024, 0, stream>>>(PART, U, W, SCAL, 3, 4);
    colpass_kernel<<<cgrid, 256, 0, stream>>>(Y1, Y0, W, PART, SCAL);
    reduce_vec_kernel<<<1, 1024, 0, stream>>>(PART, W, U, SCAL, 4, 3);
  }
  // final v = b / (K^T u)   (lands in W; S_v in SCAL[4])
  colpass_kernel<<<cgrid, 256, 0, stream>>>(Y0, Y1, U, PART, SCAL);
  reduce_vec_kernel<<<1, 1024, 0, stream>>>(PART, U, W, SCAL, 3, 4);

  // 5) D = 2 * sum(T .* Mt)
  dpass_kernel<<<cgrid, 256, 0, stream>>>(Y0, Y1, U, W, DPART, SCAL);
  dfinal_kernel<<<1, 512, 0, stream>>>(DPART, SCAL, U, W, (float*)d_out);
}